// LightGCNModel_22677427323221
// MI455X (gfx1250) — compile-verified
//
#include <hip/hip_runtime.h>
#include <hip/hip_bf16.h>

#define NUSERS 256
#define NITEMS 512
#define NNODES 768
#define EMBK   64
#define BATCH  512
#define NEDGE  65536

typedef __attribute__((ext_vector_type(2))) float v2f;
typedef __attribute__((ext_vector_type(4))) float v4f;
typedef __attribute__((ext_vector_type(8))) float v8f;

__global__ void zero_f32(float* __restrict__ p, int n) {
    int t = blockIdx.x * blockDim.x + threadIdx.x;
    if (t < n) p[t] = 0.f;
}

// cnt histogram over items (BATCH adds)
__global__ void hist_items(const int* __restrict__ items, float* __restrict__ cnt) {
    int t = blockIdx.x * blockDim.x + threadIdx.x;
    if (t < BATCH) atomicAdd(&cnt[items[t]], 1.0f);
}

// degree histogram over edge destinations
__global__ void hist_deg(const int* __restrict__ edge, float* __restrict__ deg) {
    int e = blockIdx.x * blockDim.x + threadIdx.x;
    if (e < NEDGE) atomicAdd(&deg[edge[NEDGE + e]], 1.0f);
}

__global__ void dis_kernel(const float* __restrict__ deg, float* __restrict__ dis) {
    int n = blockIdx.x * blockDim.x + threadIdx.x;
    if (n < NNODES) {
        float d = deg[n];
        dis[n] = (d > 0.f) ? rsqrtf(d) : 0.f;
    }
}

__global__ void edge_w(const int* __restrict__ edge, const float* __restrict__ dis,
                       float* __restrict__ ew) {
    int e = blockIdx.x * blockDim.x + threadIdx.x;
    if (e < NEDGE) ew[e] = dis[edge[e]] * dis[edge[NEDGE + e]];
}

// x = concat(Gu, Gi); acc = x; h = x
__global__ void init_x(const float* __restrict__ Gu, const float* __restrict__ Gi,
                       float* __restrict__ acc, float* __restrict__ h) {
    int t = blockIdx.x * blockDim.x + threadIdx.x;
    if (t < NNODES * EMBK) {
        float v = (t < NUSERS * EMBK) ? Gu[t] : Gi[t - NUSERS * EMBK];
        acc[t] = v;
        h[t]   = v;
    }
}

// one block per edge, one thread per feature: hout[dst] += hin[src]*w[e]
__global__ void spmm(const int* __restrict__ edge, const float* __restrict__ ew,
                     const float* __restrict__ hin, float* __restrict__ hout) {
    int e = blockIdx.x;
    int f = threadIdx.x;
    int s = edge[e];
    int d = edge[NEDGE + e];
    atomicAdd(&hout[d * EMBK + f], hin[s * EMBK + f] * ew[e]);
}

__global__ void accum(float* __restrict__ acc, const float* __restrict__ h, float alpha) {
    int t = blockIdx.x * blockDim.x + threadIdx.x;
    if (t < NNODES * EMBK) acc[t] += h[t] * alpha;
}

// S[u,i] = sum_j side[u,i,j] * cnt[j]   -- streams side (256MB) exactly once.
// 256MB > 192MB L2, so mark the stream NON-TEMPORAL to avoid thrashing L2
// (keeps cnt / embeddings / T / gi resident for the downstream kernels).
// One wave per row of 512; b128 NT loads; cnt staged in LDS; shfl reduction.
__global__ void side_reduce(const float* __restrict__ side, const float* __restrict__ cnt,
                            float* __restrict__ S) {
    __shared__ float scnt[NITEMS];
    for (int i = threadIdx.x; i < NITEMS; i += blockDim.x) scnt[i] = cnt[i];
    __syncthreads();
    int wave = threadIdx.x >> 5;
    int lane = threadIdx.x & 31;
    size_t row = (size_t)blockIdx.x * 8 + wave;     // 0 .. 131071
    const float* p = side + row * (size_t)NITEMS;
    float sum = 0.f;
#pragma unroll
    for (int it = 0; it < 4; ++it) {
        int j = (it * 32 + lane) * 4;
        v4f v = __builtin_nontemporal_load((const v4f*)(p + j));
        sum += v.x * scnt[j] + v.y * scnt[j + 1] + v.z * scnt[j + 2] + v.w * scnt[j + 3];
    }
    for (int off = 16; off; off >>= 1) sum += __shfl_down(sum, off);
    if (lane == 0) S[row] = sum;
}

// T[b,a] = S[users[b], items[a]]
__global__ void build_T(const float* __restrict__ S, const int* __restrict__ users,
                        const int* __restrict__ items, float* __restrict__ T) {
    int t = blockIdx.x * blockDim.x + threadIdx.x;
    if (t < BATCH * BATCH) {
        int b = t >> 9, a = t & 511;
        T[t] = S[users[b] * NITEMS + items[a]];
    }
}

__global__ void gather_embeds(const float* __restrict__ acc, const int* __restrict__ users,
                              const int* __restrict__ items, float* __restrict__ gu,
                              float* __restrict__ gi) {
    int t = blockIdx.x * blockDim.x + threadIdx.x;
    if (t < BATCH * EMBK) {
        int b = t >> 6, f = t & 63;
        gu[t] = acc[users[b] * EMBK + f];
        gi[t] = acc[(NUSERS + items[b]) * EMBK + f];
    }
}

// gii = T(512x512) @ gi(512x64) via V_WMMA_F32_16X16X4_F32 (exact f32 math).
// One wave per 16x16 output tile; 32x4 = 128 tiles; K loop 512/4 = 128 WMMAs/wave.
// f32 A 16x4 frag: lanes 0-15 hold K=0,1 (v0,v1) for M=lane; lanes 16-31 hold K=2,3.
// f32 C/D: VGPR r -> M = r + 8*(lane>=16), N = lane&15.
__global__ void gemm_wmma(const float* __restrict__ T, const float* __restrict__ gi,
                          float* __restrict__ gii) {
    int lane = threadIdx.x & 31;
    int wave = threadIdx.x >> 5;
    int wid  = blockIdx.x * (blockDim.x >> 5) + wave;   // 0..127
    int tm   = wid >> 2;                                 // 0..31
    int tn   = wid & 3;                                  // 0..3
    int half = lane >> 4;
    int mn   = lane & 15;
    v8f c = {};
    const float* arow = T + (size_t)(tm * 16 + mn) * 512;
    for (int kk = 0; kk < 512; kk += 4) {
        int ka = kk + 2 * half;
        v2f a;
        a.x = arow[ka];
        a.y = arow[ka + 1];
        v2f b;
        b.x = gi[ka * EMBK + tn * 16 + mn];
        b.y = gi[(ka + 1) * EMBK + tn * 16 + mn];
        c = __builtin_amdgcn_wmma_f32_16x16x4_f32(false, a, false, b, (short)0, c,
                                                  false, false);
    }
#pragma unroll
    for (int r = 0; r < 8; ++r) {
        int row = tm * 16 + r + 8 * half;
        gii[row * EMBK + tn * 16 + mn] = c[r];
    }
}

// out[b] = dot(gu[b], gii[b]) / max(||gii[b]||, 1e-12); one wave per b
__global__ void norm_dot(const float* __restrict__ gu, const float* __restrict__ gii,
                         float* __restrict__ out) {
    int lane = threadIdx.x & 31;
    int b = blockIdx.x * (blockDim.x >> 5) + (threadIdx.x >> 5);
    float g1 = gii[b * 64 + lane], g2 = gii[b * 64 + 32 + lane];
    float u1 = gu[b * 64 + lane],  u2 = gu[b * 64 + 32 + lane];
    float ss = g1 * g1 + g2 * g2;
    float dd = u1 * g1 + u2 * g2;
    for (int off = 16; off; off >>= 1) {
        ss += __shfl_down(ss, off);
        dd += __shfl_down(dd, off);
    }
    if (lane == 0) out[b] = dd / fmaxf(sqrtf(ss), 1e-12f);
}

extern "C" void kernel_launch(void* const* d_in, const int* in_sizes, int n_in,
                              void* d_out, int out_size, void* d_ws, size_t ws_size,
                              hipStream_t stream) {
    const float* Gu    = (const float*)d_in[0];   // 256*64
    const float* Gi    = (const float*)d_in[1];   // 512*64
    const float* side  = (const float*)d_in[2];   // 256*512*512
    const int*   edge  = (const int*)d_in[3];     // 2*65536
    const int*   users = (const int*)d_in[4];     // 512
    const int*   items = (const int*)d_in[5];     // 512
    float* out = (float*)d_out;                   // 512

    // workspace carving (floats)
    float* ws  = (float*)d_ws;
    float* cnt = ws;                 // 512
    float* deg = cnt + 512;          // 768
    float* dis = deg + 768;          // 768
    float* ew  = dis + 768;          // 65536
    float* acc = ew + NEDGE;         // 49152
    float* h0  = acc + NNODES*EMBK;  // 49152
    float* h1  = h0  + NNODES*EMBK;  // 49152
    float* S   = h1  + NNODES*EMBK;  // 131072
    float* T   = S + NUSERS*NITEMS;  // 262144
    float* gu  = T + BATCH*BATCH;    // 32768
    float* gi  = gu + BATCH*EMBK;    // 32768
    float* gii = gi + BATCH*EMBK;    // 32768

    // 1) cnt + deg histograms
    zero_f32<<<(512 + 768 + 255) / 256, 256, 0, stream>>>(cnt, 512 + 768);
    hist_items<<<(BATCH + 255) / 256, 256, 0, stream>>>(items, cnt);
    hist_deg<<<NEDGE / 256, 256, 0, stream>>>(edge, deg);
    dis_kernel<<<(NNODES + 255) / 256, 256, 0, stream>>>(deg, dis);
    edge_w<<<NEDGE / 256, 256, 0, stream>>>(edge, dis, ew);

    // 2) LightGCN propagation: acc = x + h1/2 + h2/3 + h3/4
    init_x<<<(NNODES * EMBK + 255) / 256, 256, 0, stream>>>(Gu, Gi, acc, h0);
    float* hin = h0;
    float* hout = h1;
    for (int k = 0; k < 3; ++k) {
        zero_f32<<<(NNODES * EMBK + 255) / 256, 256, 0, stream>>>(hout, NNODES * EMBK);
        spmm<<<NEDGE, EMBK, 0, stream>>>(edge, ew, hin, hout);
        accum<<<(NNODES * EMBK + 255) / 256, 256, 0, stream>>>(acc, hout, 1.0f / (float)(k + 2));
        float* tmp = hin; hin = hout; hout = tmp;
    }

    // 3) S = side . cnt  (the 256MB NT stream; roofline-dominant ~11us @ 23.3TB/s)
    side_reduce<<<(NUSERS * NITEMS) / 8, 256, 0, stream>>>(side, cnt, S);

    // 4) T gather, embedding gathers
    build_T<<<(BATCH * BATCH) / 256, 256, 0, stream>>>(S, users, items, T);
    gather_embeds<<<(BATCH * EMBK) / 256, 256, 0, stream>>>(acc, users, items, gu, gi);

    // 5) gii = T @ gi via f32 WMMA (128 waves = 16 blocks x 8 waves)
    gemm_wmma<<<16, 256, 0, stream>>>(T, gi, gii);

    // 6) normalize + dot -> out (512 waves)
    norm_dot<<<BATCH / 8, 256, 0, stream>>>(gu, gii, out);
}